// TargetProbability_66030827209361
// MI455X (gfx1250) — compile-verified
//
#include <hip/hip_runtime.h>
#include <stdint.h>

// Problem constants from the reference: B=8, S=64, V=50257, D=768
#define ROWS   512          // B*S
#define VLEN   50257        // vocab
#define DMODEL 768
#define TPB    512          // 16 wave32s per block
#define NWAVE  16
#define STAGE  4096         // floats staged per block per pipeline step (16 KB)
#define WAVE_F 256          // floats per wave per step (2 x b128 per lane)
#define NBUF   3            // 3-deep async pipeline -> 2 stages always in flight
#define NSTAGE (VLEN / STAGE)        // 12 full stages -> 49152 floats
#define TAIL_BASE (NSTAGE * STAGE)   // remaining 1105 floats via direct loads

// Issue one wave's slice of stage `s` into LDS buffer `par` using the CDNA5
// async global->LDS DMA path (tracked with ASYNCcnt, not LOADcnt).
__device__ __forceinline__ void async_issue_stage(const float* __restrict__ rowBase,
                                                  int s, float (*sbuf)[STAGE],
                                                  int par, int wave, int lane) {
#pragma unroll
  for (int j = 0; j < 2; ++j) {
    const int off = wave * WAVE_F + j * 128 + lane * 4;      // float index in stage
    uint32_t laddr = (uint32_t)(uintptr_t)(&sbuf[par][off]); // LDS byte offset
    uint64_t gaddr = (uint64_t)(uintptr_t)(rowBase + (size_t)s * STAGE + off);
    asm volatile("global_load_async_to_lds_b128 %0, %1, off"
                 :
                 : "v"(laddr), "v"(gaddr)
                 : "memory");
  }
}

__global__ __launch_bounds__(TPB) void argmax_rows_kernel(
    const float* __restrict__ P,        // [ROWS, VLEN]
    float* __restrict__ idx_out_f,      // [ROWS] argmax written as float
    int* __restrict__ ws_idx,           // [ROWS] argmax (int) for gather kernel
    float* __restrict__ ws_p) {         // [ROWS] max prob for the ST scale
  __shared__ alignas(16) float sbuf[NBUF][STAGE];   // 48 KB triple buffer
  __shared__ float red_v[NWAVE];
  __shared__ int   red_i[NWAVE];

  const int row  = blockIdx.x;
  const int lane = threadIdx.x & 31;
  const int wave = threadIdx.x >> 5;
  const float* rowBase = P + (size_t)row * VLEN;

  float bv = -3.402823466e38f;   // probs are >= 0; safe identity
  int   bi = 0;

  // Prologue: prefetch stages 0 and 1 (4 async ops / wave outstanding).
  async_issue_stage(rowBase, 0, sbuf, 0, wave, lane);
  async_issue_stage(rowBase, 1, sbuf, 1, wave, lane);

  for (int s = 0; s < NSTAGE; ++s) {
    const int par = s % NBUF;
    if (s + 2 < NSTAGE) {
      // Keep two stages in flight: issue s+2, then wait until only the two
      // newest stages (4 ops) remain outstanding => stage s has landed in LDS.
      async_issue_stage(rowBase, s + 2, sbuf, (s + 2) % NBUF, wave, lane);
      asm volatile("s_wait_asynccnt 0x4" ::: "memory");
    } else if (s + 1 < NSTAGE) {
      asm volatile("s_wait_asynccnt 0x2" ::: "memory");
    } else {
      asm volatile("s_wait_asynccnt 0x0" ::: "memory");
    }

#pragma unroll
    for (int j = 0; j < 2; ++j) {
      const int off = wave * WAVE_F + j * 128 + lane * 4;
      const float4 x = *(const float4*)(&sbuf[par][off]);   // ds_load_b128
      const int gi = s * STAGE + off;
      if (x.x > bv) { bv = x.x; bi = gi;     }
      if (x.y > bv) { bv = x.y; bi = gi + 1; }
      if (x.z > bv) { bv = x.z; bi = gi + 2; }
      if (x.w > bv) { bv = x.w; bi = gi + 3; }
    }
    // WAR fence: our LDS reads of this buffer must retire before the async
    // engine may overwrite it (buffer par is re-targeted at iteration s+1's
    // prefetch of stage s+3... i.e. one full iteration later).
    asm volatile("s_wait_dscnt 0x0" ::: "memory");
  }

  // Tail (1105 floats) via plain strided global loads.
  for (int i = TAIL_BASE + threadIdx.x; i < VLEN; i += TPB) {
    const float x = rowBase[i];
    if (x > bv) { bv = x; bi = i; }
  }

  // Wave32 butterfly reduction, lowest-index tie-break (matches jnp.argmax).
#pragma unroll
  for (int m = 16; m; m >>= 1) {
    const float ov = __shfl_xor(bv, m, 32);
    const int   oi = __shfl_xor(bi, m, 32);
    if (ov > bv || (ov == bv && oi < bi)) { bv = ov; bi = oi; }
  }
  if (lane == 0) { red_v[wave] = bv; red_i[wave] = bi; }
  __syncthreads();

  if (wave == 0) {
    bv = (lane < NWAVE) ? red_v[lane] : -3.402823466e38f;
    bi = (lane < NWAVE) ? red_i[lane] : 0x7fffffff;
#pragma unroll
    for (int m = 16; m; m >>= 1) {
      const float ov = __shfl_xor(bv, m, 32);
      const int   oi = __shfl_xor(bi, m, 32);
      if (ov > bv || (ov == bv && oi < bi)) { bv = ov; bi = oi; }
    }
    if (lane == 0) {
      ws_idx[row]    = bi;
      ws_p[row]      = bv;
      idx_out_f[row] = (float)bi;   // tuple output #2, non-bf16 => float buffer
    }
  }
}

// out[row,:] = c * W[idx[row],:], c = (1 - p) + p  (exact ST forward scale)
__global__ __launch_bounds__(DMODEL / 4) void gather_scale_kernel(
    const float* __restrict__ W,     // [VLEN, DMODEL]
    const int* __restrict__ ws_idx,
    const float* __restrict__ ws_p,
    float* __restrict__ out) {       // [ROWS, DMODEL]
  const int row = blockIdx.x;
  const int t   = threadIdx.x;             // 0..191, one float4 each
  const int v   = ws_idx[row];
  const float p = ws_p[row];
  const float c = (1.0f - p) + p;
  const float4* src = (const float4*)(W + (size_t)v * DMODEL);
  float4*       dst = (float4*)(out + (size_t)row * DMODEL);
  float4 x = src[t];
  x.x *= c; x.y *= c; x.z *= c; x.w *= c;
  dst[t] = x;
}

extern "C" void kernel_launch(void* const* d_in, const int* in_sizes, int n_in,
                              void* d_out, int out_size, void* d_ws, size_t ws_size,
                              hipStream_t stream) {
  (void)in_sizes; (void)n_in; (void)out_size; (void)ws_size;
  const float* P = (const float*)d_in[0];   // pred_probs [8,64,50257] f32
  const float* W = (const float*)d_in[1];   // embed_weight [50257,768] f32

  float* out       = (float*)d_out;
  float* emb_out   = out;                              // ROWS*DMODEL floats
  float* idx_out_f = out + (size_t)ROWS * DMODEL;      // ROWS floats (indices)

  int*   ws_idx = (int*)d_ws;
  float* ws_p   = (float*)((char*)d_ws + ROWS * sizeof(int));

  argmax_rows_kernel<<<ROWS, TPB, 0, stream>>>(P, idx_out_f, ws_idx, ws_p);
  gather_scale_kernel<<<ROWS, DMODEL / 4, 0, stream>>>(W, ws_idx, ws_p, emb_out);
}